// PartialPair_11175504904265
// MI455X (gfx1250) — compile-verified
//
#include <hip/hip_runtime.h>
#include <math.h>

// ---------------------------------------------------------------------------
// PartialPair pfaffian pipeline for MI455X (gfx1250, wave32)
//   Phase 1: Ja = (J - J^T)/2                       (480x480 fp32)
//   Phase 2: T[b] = Up[b] @ Ja                      fp32 WMMA 16x16x4
//   Phase 3: F[b] = T[b] @ Up[b]^T                  fp32 WMMA 16x16x4
//   Phase 4: Parlett-Reid LTL^T pfaffian on the antisymmetric 288x288 Ffull,
//            stored as packed strict-upper triangle in LDS (165KB < 320KB),
//            1 workgroup (288 thr) per batch element. Triangle assembled
//            directly from F + U (no Ffull global buffer, no borders kernel).
// ---------------------------------------------------------------------------

typedef float v2f __attribute__((ext_vector_type(2)));
typedef float v8f __attribute__((ext_vector_type(8)));

#define NF    512   // NFMODES
#define NE    256   // selected modes
#define NU    32    // Nunpaired
#define NM    480   // NFMODES - NUNPAIRED (paired modes)
#define NB    32    // batch
#define NFULL 288   // NE + NU  (pfaffian matrix dim)
#define NTRI  (NFULL * (NFULL - 1) / 2)   // 41328 packed upper-tri elements

// ---------------- Phase 1: antisymmetrize J --------------------------------
__global__ void build_ja(const float* __restrict__ J, float* __restrict__ Ja) {
    int tid = blockIdx.x * blockDim.x + threadIdx.x;
    if (tid >= NM * NM) return;
    int i = tid / NM, j = tid % NM;
    Ja[tid] = 0.5f * (J[i * NM + j] - J[j * NM + i]);
}

// ---------------- Phase 2: T[b] = Up[b] @ Ja -------------------------------
// Up[b][m][k] = U[idx[b][m]][NU + k]   (row gather, column slice)
// One wave per 16x16 output tile; K loop in steps of 4 (fp32 WMMA).
__global__ void gemm1(const float* __restrict__ U, const float* __restrict__ Ja,
                      const int* __restrict__ idx, float* __restrict__ T) {
    const int n0   = blockIdx.x * 16;
    const int m0   = blockIdx.y * 16;
    const int b    = blockIdx.z;
    const int lane = threadIdx.x;      // 0..31
    const int mrow = lane & 15;        // A: M index / B: N index
    const int half = lane >> 4;        // selects K pair {0,1} vs {2,3}

    const float* arow = U + (size_t)idx[b * NE + m0 + mrow] * NF + NU + 2 * half;
    const float* bcol = Ja + (size_t)(2 * half) * NM + n0 + mrow;

    v8f acc = {};
    #pragma unroll 4
    for (int kb = 0; kb < NM; kb += 4) {
        v2f a = *(const v2f*)(arow + kb);
        v2f bb;
        bb.x = bcol[(size_t)kb * NM];
        bb.y = bcol[(size_t)(kb + 1) * NM];
        acc = __builtin_amdgcn_wmma_f32_16x16x4_f32(
            false, a, false, bb, (short)0, acc, false, false);
    }

    float* trow = T + (size_t)b * NE * NM;
    #pragma unroll
    for (int r = 0; r < 8; ++r) {
        int m = m0 + r + 8 * half;                 // D: M = r + 8*(lane>>4)
        trow[(size_t)m * NM + n0 + mrow] = acc[r]; // N = lane&15
    }
}

// ---------------- Phase 3: F[b] = T[b] @ Up[b]^T  (dense 256x256) ----------
// B[k][n] = Up[b][n][k] = U[idx[b][n]][NU + k]  -> contiguous float2 per lane
__global__ void gemm2(const float* __restrict__ U, const float* __restrict__ T,
                      const int* __restrict__ idx, float* __restrict__ F) {
    const int n0   = blockIdx.x * 16;
    const int m0   = blockIdx.y * 16;
    const int b    = blockIdx.z;
    const int lane = threadIdx.x;
    const int mrow = lane & 15;
    const int half = lane >> 4;

    const float* arow = T + (size_t)b * NE * NM + (size_t)(m0 + mrow) * NM + 2 * half;
    const float* brow = U + (size_t)idx[b * NE + n0 + mrow] * NF + NU + 2 * half;

    v8f acc = {};
    #pragma unroll 4
    for (int kb = 0; kb < NM; kb += 4) {
        v2f a  = *(const v2f*)(arow + kb);
        v2f bb = *(const v2f*)(brow + kb);
        acc = __builtin_amdgcn_wmma_f32_16x16x4_f32(
            false, a, false, bb, (short)0, acc, false, false);
    }

    float* f = F + (size_t)b * NE * NE;
    #pragma unroll
    for (int r = 0; r < 8; ++r) {
        int m = m0 + r + 8 * half;
        f[(size_t)m * NE + n0 + mrow] = acc[r];
    }
}

// ---------------- Phase 4: Parlett-Reid pfaffian in LDS --------------------
// Packed strict upper triangle: elem(i,j), i<j at Tl[tbase(i)+j-i-1].
// elem(i,j) for i>j is -elem(j,i); diagonal is 0 (not stored).

__device__ __forceinline__ int tbase(int i) {
    return (i * (2 * NFULL - 1 - i)) >> 1;       // i*(575-i)/2
}
__device__ __forceinline__ int tidx(int i, int j) {  // requires i < j
    return tbase(i) + j - i - 1;
}
__device__ __forceinline__ float ldElem(const float* Tl, int i, int j) {
    if (i == j) return 0.0f;
    return (i < j) ? Tl[tidx(i, j)] : -Tl[tidx(j, i)];
}
__device__ __forceinline__ void stElem(float* Tl, int i, int j, float v) {
    if (i < j) Tl[tidx(i, j)] = v;
    else       Tl[tidx(j, i)] = -v;
}

__global__ void pfaffian(const float* __restrict__ F, const float* __restrict__ U,
                         const int* __restrict__ idx, float* __restrict__ out) {
    const int n = NFULL;
    const int b = blockIdx.x;
    const int t = threadIdx.x;     // 0..287

    __shared__ float Tl[NTRI];     // 165KB packed triangle (CDNA5: 320KB/WG)
    __shared__ float sval[512];
    __shared__ int   sidx[512];
    __shared__ float stau[NFULL];
    __shared__ float sc[NFULL];

    // ---- assemble upper triangle of Ffull = [[F, Uun], [-Uun^T, 0]] -------
    // row t entries (t, j), j = t+1..n-1
    if (t < NE) {
        const float* frow = F + (size_t)b * NE * NE + (size_t)t * NE;
        const float* urow = U + (size_t)idx[b * NE + t] * NF;
        int base = tbase(t) - t - 1;
        for (int j = t + 1; j < NE; ++j)    Tl[base + j] = frow[j];
        for (int j = NE; j < n; ++j)        Tl[base + j] = urow[j - NE];
    } else {
        int base = tbase(t) - t - 1;
        for (int j = t + 1; j < n; ++j)     Tl[base + j] = 0.0f;  // O block
    }
    __syncthreads();

    float sign = 1.0f, logabs = 0.0f;

    for (int k = 0; k < n - 1; k += 2) {
        // ---- pivot search: argmax_{i>k} |elem(i,k)| = |elem(k,i)| ----------
        sval[t] = (t > k) ? fabsf(Tl[tbase(k) + t - k - 1]) : -1.0f;
        sidx[t] = t;
        if (t < 512 - NFULL) { sval[NFULL + t] = -1.0f; sidx[NFULL + t] = 0; }
        __syncthreads();
        #pragma unroll
        for (int s = 256; s > 0; s >>= 1) {
            if (t < s && sval[t + s] > sval[t]) {
                sval[t] = sval[t + s];
                sidx[t] = sidx[t + s];
            }
            __syncthreads();
        }
        const int kp = sidx[0];    // uniform across block; kp > k

        // ---- symmetric row/col swap (k+1 <-> kp) on the triangle -----------
        if (kp != k + 1) {
            if (t != k + 1 && t != kp) {
                float v1 = ldElem(Tl, k + 1, t);
                float v2 = ldElem(Tl, kp,    t);
                stElem(Tl, k + 1, t, v2);
                stElem(Tl, kp,    t, v1);
            } else if (t == kp) {
                // pair element: new elem(k+1,kp) = old elem(kp,k+1) = -old
                Tl[tidx(k + 1, kp)] = -Tl[tidx(k + 1, kp)];
            }
        }
        __syncthreads();

        // ---- tau / c snapshot ---------------------------------------------
        const float pivot = Tl[tidx(k, k + 1)];
        stau[t] = (t > k + 1) ? Tl[tidx(k, t)] / pivot : 0.0f;   // elem(k,t)
        sc[t]   = (t > k + 1) ? -Tl[tidx(k + 1, t)]    : 0.0f;   // elem(t,k+1)
        __syncthreads();

        if (t == 0) {
            if (kp != k + 1) sign = -sign;
            sign *= (pivot > 0.0f) ? 1.0f : ((pivot < 0.0f) ? -1.0f : 0.0f);
            logabs += logf(fabsf(pivot));
        }

        // ---- rank-2 update: A[i][j] += tau[i]c[j] - c[i]tau[j], upper only -
        // parity-balanced: entry (i,j) i<j -> thread j if (i+j) even (col
        // sweep), thread i if (i+j) odd (row sweep).
        const float tt = stau[t], ct = sc[t];
        {   // column sweep: (i, t), i in [k+2, t), (i+t) even
            int i0 = k + 2;
            if ((i0 + t) & 1) i0++;
            for (int i = i0; i < t; i += 2)
                Tl[tbase(i) + t - i - 1] += stau[i] * ct - sc[i] * tt;
        }
        if (t >= k + 2) {  // row sweep: (t, j), j in (t, n), (t+j) odd
            int rb = tbase(t) - t - 1;
            for (int j = t + 1; j < n; j += 2)   // j=t+1 gives odd parity
                Tl[rb + j] += tt * sc[j] - ct * stau[j];
        }
        __syncthreads();
    }

    if (t == 0) {
        out[b]      = sign;    // signs:   out[0..31]
        out[NB + b] = logabs;  // logabs:  out[32..63]
    }
}

// ---------------------------------------------------------------------------
extern "C" void kernel_launch(void* const* d_in, const int* in_sizes, int n_in,
                              void* d_out, int out_size, void* d_ws, size_t ws_size,
                              hipStream_t stream) {
    const float* U   = (const float*)d_in[0];   // 512*512 f32
    const float* J   = (const float*)d_in[1];   // 480*480 f32
    const int*   idx = (const int*)d_in[2];     // 32*256 i32
    float* out = (float*)d_out;                 // 64 f32: sign(32) ++ logabs(32)

    // workspace partition (all fp32):
    float* Ja = (float*)d_ws;                       // 480*480
    float* T  = Ja + (size_t)NM * NM;               // 32*256*480
    float* F  = T  + (size_t)NB * NE * NM;          // 32*256*256

    build_ja<<<(NM * NM + 255) / 256, 256, 0, stream>>>(J, Ja);
    gemm1  <<<dim3(NM / 16, NE / 16, NB), 32, 0, stream>>>(U, Ja, idx, T);
    gemm2  <<<dim3(NE / 16, NE / 16, NB), 32, 0, stream>>>(U, T, idx, F);
    pfaffian<<<NB, NFULL, 0, stream>>>(F, U, idx, out);
}